// ObsOpt_50792283242814
// MI455X (gfx1250) — compile-verified
//
#include <hip/hip_runtime.h>
#include <math.h>

#define N_OBS_C 25
#define DN_STRIDE 32            // padded row: 128 B; b128-aligned halves
#define N_CON_C 31
#define IPM_ITERS_C 30
#define BIG_C 100000000.0f

typedef __attribute__((ext_vector_type(2))) float v2f;
typedef __attribute__((ext_vector_type(8))) float v8f;

// Fast hardware approximations (~1 ulp): single v_rcp_f32 / v_sqrt_f32.
__device__ __forceinline__ float frcp(float x)  { return __builtin_amdgcn_rcpf(x); }
__device__ __forceinline__ float fsqrt(float x) { return __builtin_amdgcn_sqrtf(x); }

// ---------------------------------------------------------------------------
// WMMA pre-pass: dot[i][k] = x_i . obs_k  via V_WMMA_F32_16X16X4_F32.
// Transposed mapping: A rows (M) = obstacles, B cols (N) = elements, K=4
// (ox,oy,0,0).  D layout => lane holds M = r + 8*(lane>=16), N = lane&15:
// 8 CONSECUTIVE obstacle slots for one element -> two aligned b128 stores,
// no shuffles, no per-value guards.  Two M-tiles cover 25 obstacles.
// dn itself is finished in the IPM kernel (it has ||x||^2 and ||o||^2 free).
// ---------------------------------------------------------------------------
__global__ __launch_bounds__(256)
void dot_wmma_kernel(const float* __restrict__ xc,
                     const float* __restrict__ xo,
                     float* __restrict__ dot_out, int n) {
  const int lane   = threadIdx.x & 31;
  const int wave   = threadIdx.x >> 5;
  const int groups = (n + 15) >> 4;
  const int g      = blockIdx.x * (blockDim.x >> 5) + wave;
  if (g >= groups) return;               // wave-uniform: EXEC stays all-1s
  const int  base = g << 4;
  const int  l    = lane & 15;
  const bool hi   = lane >= 16;

  int elem = base + l;
  int ei   = elem < n ? elem : n - 1;                  // clamped load
  const float2 p = reinterpret_cast<const float2*>(xc)[ei];

  v2f B;                                 // element column: K0,K1 low lanes
  B.x = hi ? 0.f : p.x;
  B.y = hi ? 0.f : p.y;

#pragma unroll
  for (int t = 0; t < 2; ++t) {
    const int j = t * 16 + l;                          // obstacle row
    float2 o = make_float2(0.f, 0.f);
    if (j < N_OBS_C) o = reinterpret_cast<const float2*>(xo)[j];
    v2f A;
    A.x = hi ? 0.f : o.x;
    A.y = hi ? 0.f : o.y;
    v8f c = {};
    c = __builtin_amdgcn_wmma_f32_16x16x4_f32(false, A, false, B,
                                              (short)0, c, false, false);
    if (elem < n) {                       // divergent only in tail group
      float4 lo4 = make_float4(c[0], c[1], c[2], c[3]);
      float4 hi4 = make_float4(c[4], c[5], c[6], c[7]);
      float4* dst = reinterpret_cast<float4*>(
          dot_out + (size_t)elem * DN_STRIDE + t * 16 + (hi ? 8 : 0));
      dst[0] = lo4;                       // obstacles M .. M+3
      dst[1] = hi4;                       // obstacles M+4 .. M+7
    }
  }
}

// ---------------------------------------------------------------------------
// Main IPM kernel: one thread = one batch element.  Structured constraint
// rows (clf / 25 two-nonzero cbf rows / 4 unit rows / delta row), symmetric
// adjugate 3x3 solve, state fully in registers via constant-index unrolling.
// __launch_bounds__(256) raises the per-wave VGPR budget (vs the default
// 1024-thread assumption) so the ~160-register IPM state does not spill.
// ---------------------------------------------------------------------------
__global__ __launch_bounds__(256)
void ipm_kernel(const float* __restrict__ xc, const float* __restrict__ xt,
                const float* __restrict__ xo, const float* __restrict__ ro,
                const float* __restrict__ lbdp, const float* __restrict__ alphap,
                const float* __restrict__ dot_pre,   // may be nullptr
                float* __restrict__ out, float* __restrict__ partial, int n) {
  __shared__ float red[256];
  const int i = blockIdx.x * blockDim.x + threadIdx.x;
  float closs = 0.f;

  if (i < n) {
    const float lbd   = lbdp[0];
    const float alpha = alphap[0];
    const float2 pc = reinterpret_cast<const float2*>(xc)[i];
    const float2 pt = reinterpret_cast<const float2*>(xt)[i];
    const float dxt = pc.x - pt.x, dyt = pc.y - pt.y;
    const float a0x = 2.f * dxt, a0y = 2.f * dyt;
    const float h0  = -lbd * (dxt * dxt + dyt * dyt);
    const float xx  = pc.x * pc.x + pc.y * pc.y;       // ||x||^2

    // staged WMMA dot products: 7 aligned b128 loads cover k = 0..24
    float dotv[28];
    const bool use_pre = (dot_pre != nullptr);
    if (use_pre) {
      const float4* row = reinterpret_cast<const float4*>(dot_pre + (size_t)i * DN_STRIDE);
#pragma unroll
      for (int q = 0; q < 7; ++q) {
        const float4 v = row[q];
        dotv[4 * q + 0] = v.x; dotv[4 * q + 1] = v.y;
        dotv[4 * q + 2] = v.z; dotv[4 * q + 3] = v.w;
      }
    }

    // cbf rows (already negated: g_k = -dir_k) and rhs
    float gkx[N_OBS_C], gky[N_OBS_C], hk[N_OBS_C];
#pragma unroll
    for (int k = 0; k < N_OBS_C; ++k) {
      const float2 o = reinterpret_cast<const float2*>(xo)[k];
      const float ddx = pc.x - o.x, ddy = pc.y - o.y;
      float dn;
      if (use_pre) {
        const float oo = o.x * o.x + o.y * o.y;
        dn = fsqrt(fmaxf(xx + oo - 2.f * dotv[k], 0.f));
      } else {
        dn = fsqrt(ddx * ddx + ddy * ddy);
      }
      const float inv = frcp(dn + 1e-6f);
      gkx[k] = -ddx * inv;
      gky[k] = -ddy * inv;
      hk[k]  = alpha * (dn - ro[k]);
      closs += fmaxf(ro[k] - dn, 0.f);
    }

    float lam[N_CON_C], s[N_CON_C];
#pragma unroll
    for (int j = 0; j < N_CON_C; ++j) { lam[j] = 1.f; s[j] = 1.f; }
    float z0 = 0.f, z1 = 0.f, z2 = 0.f;

#pragma unroll 1
    for (int it = 0; it < IPM_ITERS_C; ++it) {
      // mu
      float sumls = 0.f;
#pragma unroll
      for (int j = 0; j < N_CON_C; ++j) sumls += lam[j] * s[j];
      const float smu = 0.1f * (sumls * (1.f / 31.f));

      // M = Q + G^T D G ;  rhs = -Qz + G^T (w - lam)
      float M00 = 1.f, M11 = 1.f, M22 = 0.05f, M01 = 0.f, M02 = 0.f, M12 = 0.f;
      float r0 = -z0, r1 = -z1, r2 = -0.05f * z2;

      auto accRow = [&](float gx, float gy, float gz, float hj, float lj, float sj) {
        const float Gz   = gx * z0 + gy * z1 + gz * z2;
        const float rp   = Gz + sj - hj;
        const float invs = frcp(sj);
        const float D    = lj * invs;
        const float rc   = lj * sj - smu;
        const float w    = (rc - lj * rp) * invs;
        const float cc   = w - lj;
        r0 += cc * gx; r1 += cc * gy; r2 += cc * gz;
        M00 += D * gx * gx; M01 += D * gx * gy; M02 += D * gx * gz;
        M11 += D * gy * gy; M12 += D * gy * gz; M22 += D * gz * gz;
      };
      accRow(a0x, a0y, -1.f, h0, lam[0], s[0]);
#pragma unroll
      for (int k = 0; k < N_OBS_C; ++k)
        accRow(gkx[k], gky[k], 0.f, hk[k], lam[k + 1], s[k + 1]);
      accRow( 1.f, 0.f, 0.f, 0.05f, lam[26], s[26]);
      accRow(-1.f, 0.f, 0.f, 0.05f, lam[27], s[27]);
      accRow(0.f,  1.f, 0.f, 0.05f, lam[28], s[28]);
      accRow(0.f, -1.f, 0.f, 0.05f, lam[29], s[29]);
      accRow(0.f, 0.f, -1.f, 0.f,   lam[30], s[30]);

      // symmetric 3x3 solve by adjugate (M is SPD)
      const float A00 = M11 * M22 - M12 * M12;
      const float A01 = M02 * M12 - M01 * M22;
      const float A02 = M01 * M12 - M02 * M11;
      const float A11 = M00 * M22 - M02 * M02;
      const float A12 = M01 * M02 - M00 * M12;
      const float A22 = M00 * M11 - M01 * M01;
      const float det  = M00 * A00 + M01 * A01 + M02 * A02;
      const float idet = frcp(det);
      const float dz0 = (A00 * r0 + A01 * r1 + A02 * r2) * idet;
      const float dz1 = (A01 * r0 + A11 * r1 + A12 * r2) * idet;
      const float dz2 = (A02 * r0 + A12 * r1 + A22 * r2) * idet;

      // per-row direction (recomputed instead of stored: saves 62 VGPRs)
      float amin = BIG_C;
      float a = 0.f;                  // set between the two phases
      auto dirRow = [&](float gx, float gy, float gz, float hj,
                        float& lj, float& sj, bool update) {
        const float Gz  = gx * z0 + gy * z1 + gz * z2;
        const float rp  = Gz + sj - hj;
        const float rc  = lj * sj - smu;
        const float gdz = gx * dz0 + gy * dz1 + gz * dz2;
        const float dsj = -rp - gdz;
        const float dlj = (-rc - lj * dsj) * frcp(sj);
        if (!update) {
          if (dlj < 0.f) amin = fminf(amin, -lj * frcp(dlj));
          if (dsj < 0.f) amin = fminf(amin, -sj * frcp(dsj));
        } else {
          lj += a * dlj;
          sj += a * dsj;
        }
      };

      auto allRows = [&](bool update) {
        dirRow(a0x, a0y, -1.f, h0, lam[0], s[0], update);
#pragma unroll
        for (int k = 0; k < N_OBS_C; ++k)
          dirRow(gkx[k], gky[k], 0.f, hk[k], lam[k + 1], s[k + 1], update);
        dirRow( 1.f, 0.f, 0.f, 0.05f, lam[26], s[26], update);
        dirRow(-1.f, 0.f, 0.f, 0.05f, lam[27], s[27], update);
        dirRow(0.f,  1.f, 0.f, 0.05f, lam[28], s[28], update);
        dirRow(0.f, -1.f, 0.f, 0.05f, lam[29], s[29], update);
        dirRow(0.f, 0.f, -1.f, 0.f,   lam[30], s[30], update);
      };

      allRows(false);                       // phase 1: step length
      a = fminf(1.f, 0.99f * amin);
      allRows(true);                        // phase 2: update lam, s (old z)
      z0 += a * dz0; z1 += a * dz1; z2 += a * dz2;
    }

    out[2 * i]     = z0;
    out[2 * i + 1] = z1;
  }

  // deterministic fixed-order block reduction for collision loss
  red[threadIdx.x] = closs;
  __syncthreads();
  for (int off = blockDim.x >> 1; off > 0; off >>= 1) {
    if ((int)threadIdx.x < off) red[threadIdx.x] += red[threadIdx.x + off];
    __syncthreads();
  }
  if (threadIdx.x == 0) partial[blockIdx.x] = red[0];
}

__global__ __launch_bounds__(256)
void reduce_kernel(const float* __restrict__ partial, int nblocks,
                   float* __restrict__ out_loss) {
  __shared__ float red[256];
  float v = 0.f;
  for (int idx = threadIdx.x; idx < nblocks; idx += 256) v += partial[idx];
  red[threadIdx.x] = v;
  __syncthreads();
  for (int off = 128; off > 0; off >>= 1) {
    if ((int)threadIdx.x < off) red[threadIdx.x] += red[threadIdx.x + off];
    __syncthreads();
  }
  if (threadIdx.x == 0) *out_loss = red[0];
}

extern "C" void kernel_launch(void* const* d_in, const int* in_sizes, int n_in,
                              void* d_out, int out_size, void* d_ws, size_t ws_size,
                              hipStream_t stream) {
  const float* xc    = (const float*)d_in[0];
  const float* xt    = (const float*)d_in[1];
  const float* xo    = (const float*)d_in[2];
  const float* ro    = (const float*)d_in[3];
  const float* lbd   = (const float*)d_in[4];
  const float* alpha = (const float*)d_in[5];
  const int n = in_sizes[0] / 2;

  float* out = (float*)d_out;
  const int nblocks = (n + 255) / 256;

  const size_t dot_bytes = (size_t)n * DN_STRIDE * sizeof(float);
  float* dot_ptr = nullptr;
  float* partial = (float*)d_ws;

  if (ws_size >= dot_bytes + (size_t)nblocks * sizeof(float) + 256) {
    dot_ptr = (float*)d_ws;
    partial = (float*)((char*)d_ws + dot_bytes);
    const int groups  = (n + 15) / 16;
    const int pblocks = (groups + 7) / 8;   // 8 waves (tiles) per 256-thr block
    dot_wmma_kernel<<<pblocks, 256, 0, stream>>>(xc, xo, dot_ptr, n);
  }

  ipm_kernel<<<nblocks, 256, 0, stream>>>(xc, xt, xo, ro, lbd, alpha,
                                          dot_ptr, out, partial, n);
  reduce_kernel<<<1, 256, 0, stream>>>(partial, nblocks, out + (size_t)2 * n);
}